// PlenoxelModel_919123002047
// MI455X (gfx1250) — compile-verified
//
#include <hip/hip_runtime.h>
#include <math.h>

// Plenoxel renderer for MI455X (gfx1250).
//
// Strategy (roofline-driven):
//  1) SH basis is constant per launch and interpolation/projection commute,
//     so fold the 27 SH coeffs into RGB per *voxel* first:
//        g4[v] = (sigma, R, G, B) = V[v,0:28] x P[28,4]
//     Real GEMM -> V_WMMA_F32_16X16X4_F32 (exact f32). Wave-persistent:
//     B fragments (from SH basis) hoisted once per thread, then each wave
//     streams several 16-voxel tiles (7 b64 loads + 7 WMMAs per tile).
//     Output grid is 128^3 * 16B = 33.5 MB -> fits in the 192 MB L2.
//  2) Ray-march gathers one float4 (global_load_b128) per corner from the
//     L2-resident reduced grid, fully branchless (in-bounds flags folded
//     into per-axis weights, clamped indices, unconditional gathers).

#define GXD 128
#define GYD 128
#define GZD 128
#define VOXDIM 28
#define NSAMP 64
#define DTCONST 2.0f        // (T_FAR - T_NEAR) / NUM_SAMPLES = 128/64
#define NTILES (GXD * GYD * GZD / 16)  // 131072 tiles of 16 voxels

typedef float v2f __attribute__((ext_vector_type(2)));
typedef float v8f __attribute__((ext_vector_type(8)));

// P[k][n] for the projection GEMM  out[M,4] = V[M,28] x P.
//   column 0: sigma passthrough (P[0][0] = 1)
//   columns 1..3: SH basis for R,G,B  (P[1+9c+q][1+c] = sh[q])
__device__ __forceinline__ float Pval(int k, int n, const float* sh) {
  if (n == 0) return (k == 0) ? 1.0f : 0.0f;
  if (n > 3) return 0.0f;
  int q = k - 1 - 9 * (n - 1);
  return (q >= 0 && q < 9) ? sh[q] : 0.0f;
}

// ---------------------------------------------------------------------------
// Pass 1: per-voxel SH projection via f32 WMMA, wave-persistent.
//
// ISA 7.12.2 operand layouts (wave32):
//   A 16x4 f32 : lane = m + 16*(k>>1), vgpr = k&1
//   B 4x16 f32 : lane = n + 16*(k>>1), vgpr = k&1   (mirror of A, m<->n)
//   C/D 16x16  : lane = n + 16*(M>=8), vgpr = M&7
// ---------------------------------------------------------------------------
__global__ void __launch_bounds__(256) plenoxel_project_wmma(
    const float* __restrict__ vox,   // [128^3][28]
    const float* __restrict__ ang,   // [2] theta, phi
    float* __restrict__ g4,          // [128^3][4]
    int nwaves) {                    // total waves in launch
  const int lane = threadIdx.x & 31;
  const int wave = threadIdx.x >> 5;
  const int m    = lane & 15;       // A row / B col / D col
  const int hi   = lane >> 4;       // selects K pair (A/B) and D row half
  const int wid  = blockIdx.x * (blockDim.x >> 5) + wave;

  // Degree-2 real SH basis, same coefficient order as the reference.
  const float th = ang[0], ph = ang[1];
  const float st = sinf(th), ct = cosf(th);
  const float sp = sinf(ph), cp = cosf(ph);
  float sh[9];
  sh[0] = 0.28209479177387814f;                 // 0.5*sqrt(1/pi)
  sh[1] = 0.4886025119029199f * st * sp;        // 0.5*sqrt(3/pi)
  sh[2] = 0.4886025119029199f * ct;
  sh[3] = 0.4886025119029199f * st * cp;
  sh[4] = 1.0925484305920792f * (st * cp) * (st * sp);    // 0.5*sqrt(15/pi)
  sh[5] = 1.0925484305920792f * (st * sp) * ct;
  sh[6] = 0.31539156525252005f * (3.0f * ct * ct - 1.0f); // 0.25*sqrt(5/pi)
  sh[7] = 1.0925484305920792f * (st * cp) * ct;
  sh[8] = 0.5462742152960396f *                           // 0.25*sqrt(15/pi)
          ((st * cp) * (st * cp) - (st * sp) * (st * sp));

  // Hoist B fragments: loop-invariant across tiles (depend only on lane).
  float bx[7], by[7];
#pragma unroll
  for (int c = 0; c < 7; ++c) {
    const int k0 = 4 * c + 2 * hi;
    bx[c] = Pval(k0, m, sh);
    by[c] = Pval(k0 + 1, m, sh);
  }

  for (int tile = wid; tile < NTILES; tile += nwaves) {
    const long long base = (long long)tile * 16;
    const float* ap0 = vox + (base + m) * VOXDIM;

    v8f acc = {};
#pragma unroll
    for (int c = 0; c < 7; ++c) {        // K = 28 = 7 chunks of 4
      const int k0 = 4 * c + 2 * hi;
      v2f a, b;
      a.x = ap0[k0];
      a.y = ap0[k0 + 1];
      b.x = bx[c];
      b.y = by[c];
      // 8 args: (neg_a, A, neg_b, B, c_mod, C, reuse_a, reuse_b)
      acc = __builtin_amdgcn_wmma_f32_16x16x4_f32(
          false, a, false, b, (short)0, acc, false, false);
    }

    // D columns 0..3 hold (sigma,R,G,B); lane holds rows M = v + 8*hi.
    if (m < 4) {
#pragma unroll
      for (int v = 0; v < 8; ++v) {
        g4[(base + v + 8 * hi) * 4 + m] = acc[v];
      }
    }
  }
}

// ---------------------------------------------------------------------------
// Pass 2: ray march over the L2-resident (sigma,R,G,B) grid. Branchless
// trilinear gather; inclusive-cumsum transmittance exactly as reference.
// ---------------------------------------------------------------------------
__global__ void __launch_bounds__(256) plenoxel_raymarch(
    const float4* __restrict__ g4,   // [128^3] (sigma,R,G,B)
    const float* __restrict__ ro,    // [R][3]
    const float* __restrict__ rd,    // [R][3]
    float* __restrict__ out,         // [R][3]
    int nrays) {
  const int r = blockIdx.x * blockDim.x + threadIdx.x;
  if (r >= nrays) return;

  const float ox = ro[3 * r + 0], oy = ro[3 * r + 1], oz = ro[3 * r + 2];
  const float dx = rd[3 * r + 0], dy = rd[3 * r + 1], dz = rd[3 * r + 2];

  float trans = 0.0f, aR = 0.0f, aG = 0.0f, aB = 0.0f;

  for (int s = 0; s < NSAMP; ++s) {
    const float t = ((float)s + 0.5f) * DTCONST;
    const float px = fmaf(t, dx, ox);
    const float py = fmaf(t, dy, oy);
    const float pz = fmaf(t, dz, oz);
    const float fx = floorf(px), fy = floorf(py), fz = floorf(pz);
    const int ix = (int)fx, iy = (int)fy, iz = (int)fz;
    const float rx = px - fx, ry = py - fy, rz = pz - fz;

    // Per-axis weights with the in-bounds flag folded in (out-of-range
    // corner => zero weight, exactly the reference's `inside` mask), plus
    // clamped, pre-shifted indices so all 8 gathers are unconditional.
    float wxf[2], wyf[2], wzf[2];
    int xo[2], yo[2], zo[2];
#pragma unroll
    for (int c = 0; c < 2; ++c) {
      const int X = ix + c, Y = iy + c, Z = iz + c;
      const float wx = c ? rx : 1.0f - rx;
      const float wy = c ? ry : 1.0f - ry;
      const float wz = c ? rz : 1.0f - rz;
      wxf[c] = ((unsigned)X < (unsigned)GXD) ? wx : 0.0f;
      wyf[c] = ((unsigned)Y < (unsigned)GYD) ? wy : 0.0f;
      wzf[c] = ((unsigned)Z < (unsigned)GZD) ? wz : 0.0f;
      const int Xc = X < 0 ? 0 : (X > GXD - 1 ? GXD - 1 : X);
      const int Yc = Y < 0 ? 0 : (Y > GYD - 1 ? GYD - 1 : Y);
      const int Zc = Z < 0 ? 0 : (Z > GZD - 1 ? GZD - 1 : Z);
      xo[c] = Xc << 14;   // * 128*128
      yo[c] = Yc << 7;    // * 128
      zo[c] = Zc;
    }

    float sig = 0.0f, cr = 0.0f, cg = 0.0f, cb = 0.0f;
#pragma unroll
    for (int cx = 0; cx < 2; ++cx) {
#pragma unroll
      for (int cy = 0; cy < 2; ++cy) {
        const float wxy = wxf[cx] * wyf[cy];
        const int oxy = xo[cx] + yo[cy];
#pragma unroll
        for (int cz = 0; cz < 2; ++cz) {
          const float w = wxy * wzf[cz];
          const float4 v = g4[oxy + zo[cz]];   // unconditional b128 gather
          sig = fmaf(w, v.x, sig);
          cr  = fmaf(w, v.y, cr);
          cg  = fmaf(w, v.z, cg);
          cb  = fmaf(w, v.w, cb);
        }
      }
    }

    // Reference: exp_term = exp(-sigma*DT); transmittance = inclusive cumsum.
    const float e = expf(-sig * DTCONST);
    trans += e;                       // outside-grid samples add exp(0)=1
    const float wgt = trans * (1.0f - e);
    aR = fmaf(wgt, cr, aR);
    aG = fmaf(wgt, cg, aG);
    aB = fmaf(wgt, cb, aB);
  }

  out[3 * r + 0] = aR;
  out[3 * r + 1] = aG;
  out[3 * r + 2] = aB;
}

// ---------------------------------------------------------------------------
extern "C" void kernel_launch(void* const* d_in, const int* in_sizes, int n_in,
                              void* d_out, int out_size, void* d_ws,
                              size_t ws_size, hipStream_t stream) {
  (void)n_in; (void)out_size; (void)ws_size;
  const float* vox = (const float*)d_in[0];   // [128,128,128,28] f32
  const float* ro  = (const float*)d_in[1];   // [R,3] f32
  const float* rd  = (const float*)d_in[2];   // [R,3] f32
  const float* ang = (const float*)d_in[3];   // [2] f32
  float* out = (float*)d_out;                 // [R,3] f32

  // Workspace: reduced grid, 128^3 * 4 floats = 33,554,432 bytes.
  float* g4 = (float*)d_ws;

  const int nrays = in_sizes[1] / 3;          // 65,536

  // Pass 1: wave-persistent projection. 2048 blocks x 8 waves = 16384 waves,
  // each streams 8 tiles (131072 tiles total).
  const int pblocks = 2048;
  const int pwaves  = pblocks * (256 / 32);
  plenoxel_project_wmma<<<pblocks, 256, 0, stream>>>(vox, ang, g4, pwaves);

  // Pass 2: one thread per ray.
  plenoxel_raymarch<<<(nrays + 255) / 256, 256, 0, stream>>>(
      (const float4*)g4, ro, rd, out, nrays);
}